// GeometricSpatialRelationLoss_70248485093937
// MI455X (gfx1250) — compile-verified
//
#include <hip/hip_runtime.h>

// GeometricSpatialRelationLoss for MI455X (gfx1250), wave32 + WMMA.
//
// Restructure: the appended "extra" node in each gather row duplicates an
// existing node -> it cannot change any min, only double-weights that node's
// points in the sum. So:
//   phase 1 (per direction): minNode[i][m] = min_{j in node m} d2(p_i, q_j),
//       one 8960x8960 pass on V_WMMA_F32_16X16X4_F32 instead of 26 replicas.
//   phase 2: per point, min/argmin/2nd-min over the 28 node-mins gives every
//       "min over nodes != idx" in O(1); duplicate node handled by weight 2;
//       half-arch chamfers are mins over node ranges of the same table.
//
// WMMA folding: with K=3 padded to 4, the whole squared-distance formula sits
// inside the matrix op:
//   A row = [-2x0,-2x1,-2x2, 1], B col = [y0,y1,y2,|y|^2], C = |x|^2 per row
//   => D = |x|^2 + |y|^2 - 2 x.y, no epilogue FMA/ADD, only running mins.
//
// Parallelism: node axis split 4 ways per row-strip (disjoint outputs) ->
// 2240 waves; tile loop unrolled 2x for two independent WMMAs in flight.

typedef __attribute__((ext_vector_type(2))) float v2f;
typedef __attribute__((ext_vector_type(8))) float v8f;

#define NODES  28
#define PPN    320
#define NPTS   (NODES * PPN)   // 8960
#define MID    13              // (NODES-2)/2
#define NSPLIT 4               // node-axis split per row strip (28/4 = 7)
#define FINF   3.402823466e38f

__device__ __forceinline__ float clip10(float v) {
  return fminf(fmaxf(v, -10.0f), 10.0f);
}

// One wave handles a 16-row strip of A against 7 nodes of B, keeping per-node
// running row-mins. 32b A 16x4 layout: VGPR0 = K0 (lanes 0-15) / K2
// (lanes 16-31), VGPR1 = K1 / K3. B 4x16: rows K striped across lane halves.
// C/D: lane l holds rows {k + 8*(l/16)}, col l%16 in VGPR k.
__global__ __launch_bounds__(256) void chamfer_min_kernel(
    const float* __restrict__ A, const float* __restrict__ B,
    float* __restrict__ minNode) {
  const int lane    = threadIdx.x & 31;
  const int wave    = blockIdx.x * (blockDim.x >> 5) + (threadIdx.x >> 5);
  const int rowTile = wave >> 2;              // 0..559
  const int mBase   = (wave & 3) * (NODES / NSPLIT);  // 0,7,14,21
  const int rowBase = rowTile * 16;
  const int row     = lane & 15;
  const int hi      = (lane >= 16) ? 8 : 0;

  // Load + clip this wave's 16 A points; build the scaled A fragment.
  const float* ap = A + (size_t)(rowBase + row) * 3;
  const float ax = clip10(ap[0]), ay = clip10(ap[1]), az = clip10(ap[2]);
  const float nx = ax * ax + ay * ay + az * az;
  v2f a;
  a.x = (lane < 16) ? (-2.0f * ax) : (-2.0f * az);  // K=0 | K=2
  a.y = (lane < 16) ? (-2.0f * ay) : 1.0f;          // K=1 | K=3 (ones column)

  // C init: |x|^2 per row, replicated across columns -> matches C/D layout.
  v8f cinit;
#pragma unroll
  for (int k = 0; k < 8; ++k) cinit[k] = __shfl(nx, k + hi, 32);

  for (int mm = 0; mm < NODES / NSPLIT; ++mm) {
    const int m = mBase + mm;
    float rmv[8];
#pragma unroll
    for (int k = 0; k < 8; ++k) rmv[k] = FINF;

    const float* bbase = B + (size_t)m * PPN * 3;
#pragma unroll 2
    for (int t = 0; t < PPN / 16; ++t) {
      const float* bp = bbase + (size_t)(t * 16 + row) * 3;
      const float bx = clip10(bp[0]), by = clip10(bp[1]), bz = clip10(bp[2]);
      const float ny = bx * bx + by * by + bz * bz;
      v2f b;
      b.x = (lane < 16) ? bx : by;   // K=0 row | K=1 row
      b.y = (lane < 16) ? bz : ny;   // K=2 row | K=3 row (|y|^2)

      // D = C + A*B = |x|^2 + |y|^2 - 2 x.y  (full squared distance)
      v8f d = __builtin_amdgcn_wmma_f32_16x16x4_f32(
          /*neg_a=*/false, a, /*neg_b=*/false, b,
          /*c_mod=*/(short)0, cinit, /*reuse_a=*/false, /*reuse_b=*/false);

#pragma unroll
      for (int k = 0; k < 8; ++k) rmv[k] = fminf(rmv[k], d[k]);
    }

    // Row-min across the 16 columns held by each lane half.
#pragma unroll
    for (int off = 8; off >= 1; off >>= 1) {
#pragma unroll
      for (int k = 0; k < 8; ++k)
        rmv[k] = fminf(rmv[k], __shfl_xor(rmv[k], off, 32));
    }
    if (lane == 0 || lane == 16) {
#pragma unroll
      for (int k = 0; k < 8; ++k)
        minNode[(size_t)(rowBase + hi + k) * NODES + m] = rmv[k];
    }
  }
}

// One thread per (point, direction): fold the 28 node-mins into this point's
// contribution to all 26 subset losses + the half-arch loss. Block-reduce.
__global__ __launch_bounds__(256) void subset_reduce_kernel(
    const float* __restrict__ minX, const float* __restrict__ minY,
    float* __restrict__ partials) {
  const int gid = blockIdx.x * blockDim.x + threadIdx.x;
  float tot = 0.0f;
  if (gid < 2 * NPTS) {
    const float* tab = (gid < NPTS) ? minX : minY;
    const int i = (gid < NPTS) ? gid : gid - NPTS;
    float v[NODES];
    const float* r = tab + (size_t)i * NODES;
#pragma unroll
    for (int m = 0; m < NODES; ++m) v[m] = r[m];

    float m1 = FINF; int a1 = 0;
#pragma unroll
    for (int m = 0; m < NODES; ++m)
      if (v[m] < m1) { m1 = v[m]; a1 = m; }
    float m2 = FINF;
#pragma unroll
    for (int m = 0; m < NODES; ++m)
      if (m != a1) m2 = fminf(m2, v[m]);

    const int n = i / PPN;  // node of this point
#pragma unroll
    for (int idx = 0; idx < NODES - 2; ++idx) {
      const float em = (a1 == idx) ? m2 : m1;      // min over nodes != idx
      const int extra = (idx < MID) ? idx + MID : idx - MID;
      const float w = ((n != idx) ? 1.0f : 0.0f) + ((extra == n) ? 1.0f : 0.0f);
      tot += w * em;
    }

    float hm = FINF;  // half-arch chamfer term for this point
    if (n < MID) {
#pragma unroll
      for (int m = 0; m < MID; ++m) hm = fminf(hm, v[m]);
    } else {
#pragma unroll
      for (int m = MID; m < NODES; ++m) hm = fminf(hm, v[m]);
    }
    tot += hm;
  }

  __shared__ float sd[256];
  sd[threadIdx.x] = tot;
  __syncthreads();
  for (int s = 128; s > 0; s >>= 1) {
    if ((int)threadIdx.x < s) sd[threadIdx.x] += sd[threadIdx.x + s];
    __syncthreads();
  }
  if (threadIdx.x == 0) partials[blockIdx.x] = sd[0];
}

__global__ __launch_bounds__(128) void final_reduce_kernel(
    const float* __restrict__ partials, int n, float* __restrict__ out) {
  float s = 0.0f;
  for (int i = threadIdx.x; i < n; i += blockDim.x) s += partials[i];
  __shared__ float sd[128];
  sd[threadIdx.x] = s;
  __syncthreads();
  for (int k = 64; k > 0; k >>= 1) {
    if ((int)threadIdx.x < k) sd[threadIdx.x] += sd[threadIdx.x + k];
    __syncthreads();
  }
  if (threadIdx.x == 0) out[0] = sd[0];
}

extern "C" void kernel_launch(void* const* d_in, const int* in_sizes, int n_in,
                              void* d_out, int out_size, void* d_ws, size_t ws_size,
                              hipStream_t stream) {
  (void)in_sizes; (void)n_in; (void)out_size; (void)ws_size;
  const float* X = (const float*)d_in[0];   // [28,320,3] f32
  const float* T = (const float*)d_in[1];   // [28,320,3] f32
  // d_in[2] = node_idx (==28, baked into constants)

  float* minX     = (float*)d_ws;                         // 8960*28 f32
  float* minY     = minX + (size_t)NPTS * NODES;          // 8960*28 f32
  float* partials = minY + (size_t)NPTS * NODES;          // 70 f32

  // Phase 1: 560 row-strips x 4 node-splits = 2240 waves, 8 waves per block.
  const int grid1 = (NPTS / 16) * NSPLIT / 8;  // 280
  chamfer_min_kernel<<<grid1, 256, 0, stream>>>(X, T, minX);
  chamfer_min_kernel<<<grid1, 256, 0, stream>>>(T, X, minY);

  // Phase 2: fold tables into the scalar loss.
  const int nb = (2 * NPTS + 255) / 256;  // 70
  subset_reduce_kernel<<<nb, 256, 0, stream>>>(minX, minY, partials);
  final_reduce_kernel<<<1, 128, 0, stream>>>(partials, nb, (float*)d_out);
}